// _RRoiAlign_30863634989451
// MI455X (gfx1250) — compile-verified
//
#include <hip/hip_runtime.h>
#include <hip/hip_bf16.h>

typedef float v2f __attribute__((ext_vector_type(2)));
typedef float v8f __attribute__((ext_vector_type(8)));

#define POOLED_H 8
#define POOLED_W 64
#define FEAT_C   256
#define FEAT_H   160
#define FEAT_W   160
#define FEAT_HW  (FEAT_H * FEAT_W)

#if defined(__has_builtin)
#  if __has_builtin(__builtin_amdgcn_wmma_f32_16x16x4_f32)
#    define HAVE_WMMA_F32X4 1
#  endif
#endif
#ifndef HAVE_WMMA_F32X4
#  define HAVE_WMMA_F32X4 0
#endif

__global__ __launch_bounds__(256) void rroi_align_kernel(
    const float* __restrict__ feat,
    const float* __restrict__ rois,
    float* __restrict__ out)
{
    __shared__ float s_xy[POOLED_W * 2];   // (x,y) per pw
    __shared__ int   s_off[4][POOLED_W];   // clamped flat offsets per tap
    __shared__ float s_wgt[4][POOLED_W];   // validity-masked bilinear weights

    const int bx  = blockIdx.x;
    const int n   = bx >> 3;               // roi index
    const int ph  = bx & 7;                // pooled row
    const int tid = threadIdx.x;

#if HAVE_WMMA_F32X4
    // ---------------- Phase 0: wave 0 computes all 64 sample coords with one
    // fp32 WMMA (exact): D[m, 2g+c] = coord c of position pw = 16g + m.
    if (tid < 32) {
        const float cx = rois[n * 6 + 1];
        const float cy = rois[n * 6 + 2];
        const float rh = rois[n * 6 + 3];
        const float rw = rois[n * 6 + 4];
        const float th = rois[n * 6 + 5];
        const float sx = rw * (1.0f / POOLED_W);
        const float sy = rh * (1.0f / POOLED_H);
        const float yy = (ph + 0.5f - POOLED_H * 0.5f) * sy;
        const float ct = cosf(th);
        const float st = sinf(th);

        const int  lane = tid;
        const int  nn   = lane & 15;
        const bool lo   = lane < 16;

        // A (16x4), documented layout: VGPR0 = K0|K2, VGPR1 = K1|K3.
        // Row m: [xx_base(m), 1, 0, 0] with xx_base = (m + 0.5 - 32) * sx.
        v2f a;
        a.x = lo ? (nn + 0.5f - 32.0f) * sx : 0.0f;   // K0 (K2 = 0)
        a.y = lo ? 1.0f : 0.0f;                       // K1 (K3 = 0)

        // B (4x16), assumed mirrored layout: VGPR0 = K0|K2, VGPR1 = K1|K3.
        // Column 2g+0 (x): [ct, ct*16g*sx + st*yy + cx]
        // Column 2g+1 (y): [-st, -st*16g*sx + ct*yy + cy]
        float b0v = 0.0f, b1v = 0.0f;
        if (lo && nn < 8) {
            const float gx = 16.0f * (float)(nn >> 1) * sx;
            if (nn & 1) { b0v = -st; b1v = -st * gx + ct * yy + cy; }
            else        { b0v =  ct; b1v =  ct * gx + st * yy + cx; }
        }
        v2f b; b.x = b0v; b.y = b1v;

        v8f cacc = {};
        v8f d = __builtin_amdgcn_wmma_f32_16x16x4_f32(
            false, a, false, b, (short)0, cacc, false, false);

        // D layout: VGPR r -> (M=r, N=lane) lanes 0-15, (M=r+8, N=lane-16) lanes 16-31.
        if (nn < 8) {
            const int g  = nn >> 1;
            const int cc = nn & 1;
            const int mb = lo ? 0 : 8;
#pragma unroll
            for (int r = 0; r < 8; ++r)
                s_xy[(16 * g + mb + r) * 2 + cc] = d[r];
        }
    }
    __syncthreads();
#endif

    // ---------------- Phase 1: threads 0..63 build offsets + weights per pw.
    if (tid < POOLED_W) {
#if HAVE_WMMA_F32X4
        const float x = s_xy[tid * 2 + 0];
        const float y = s_xy[tid * 2 + 1];
#else
        const float cx = rois[n * 6 + 1];
        const float cy = rois[n * 6 + 2];
        const float rh = rois[n * 6 + 3];
        const float rw = rois[n * 6 + 4];
        const float th = rois[n * 6 + 5];
        const float sx = rw * (1.0f / POOLED_W);
        const float sy = rh * (1.0f / POOLED_H);
        const float xx = (tid + 0.5f - POOLED_W * 0.5f) * sx;
        const float yy = (ph + 0.5f - POOLED_H * 0.5f) * sy;
        const float ct = cosf(th);
        const float st = sinf(th);
        const float x  =  ct * xx + st * yy + cx;
        const float y  = -st * xx + ct * yy + cy;
#endif
        const float x0f = floorf(x);
        const float y0f = floorf(y);
        const float lx  = x - x0f;
        const float ly  = y - y0f;
        const int x0 = (int)x0f, y0 = (int)y0f;
        const int x1 = x0 + 1,  y1 = y0 + 1;

        const bool vx0 = (x0 >= 0) && (x0 < FEAT_W);
        const bool vx1 = (x1 >= 0) && (x1 < FEAT_W);
        const bool vy0 = (y0 >= 0) && (y0 < FEAT_H);
        const bool vy1 = (y1 >= 0) && (y1 < FEAT_H);

        const int cx0 = min(max(x0, 0), FEAT_W - 1);
        const int cx1 = min(max(x1, 0), FEAT_W - 1);
        const int cy0 = min(max(y0, 0), FEAT_H - 1);
        const int cy1 = min(max(y1, 0), FEAT_H - 1);

        const float w00 = (1.0f - lx) * (1.0f - ly);
        const float w01 = lx * (1.0f - ly);
        const float w10 = (1.0f - lx) * ly;
        const float w11 = lx * ly;

        s_off[0][tid] = cy0 * FEAT_W + cx0;
        s_off[1][tid] = cy0 * FEAT_W + cx1;
        s_off[2][tid] = cy1 * FEAT_W + cx0;
        s_off[3][tid] = cy1 * FEAT_W + cx1;
        s_wgt[0][tid] = (vy0 && vx0) ? w00 : 0.0f;
        s_wgt[1][tid] = (vy0 && vx1) ? w01 : 0.0f;
        s_wgt[2][tid] = (vy1 && vx0) ? w10 : 0.0f;
        s_wgt[3][tid] = (vy1 && vx1) ? w11 : 0.0f;
    }
    __syncthreads();

    // ---------------- Phase 2: gather + blend over channels.
    // Thread t handles pw = t&63, channels c0, c0+4, ... (c0 = t>>6).
    const int pw = tid & (POOLED_W - 1);
    const int c0 = tid >> 6;

    const int   o0 = s_off[0][pw], o1 = s_off[1][pw];
    const int   o2 = s_off[2][pw], o3 = s_off[3][pw];
    const float w0 = s_wgt[0][pw], w1 = s_wgt[1][pw];
    const float w2 = s_wgt[2][pw], w3 = s_wgt[3][pw];

    // batch index: uniform scalar load (rois[n*6] is wave-uniform address)
    const int batch = (int)rois[n * 6 + 0];

    const float* bp = feat + ((size_t)batch * FEAT_C + (size_t)c0) * FEAT_HW;
    float* op = out + (((size_t)n * FEAT_C + (size_t)c0) * POOLED_H + (size_t)ph) * POOLED_W + pw;

#pragma unroll 4
    for (int c = c0; c < FEAT_C; c += 4) {
        const float v = w0 * bp[o0] + w1 * bp[o1] + w2 * bp[o2] + w3 * bp[o3];
        *op = v;
        bp += 4 * (size_t)FEAT_HW;
        op += 4 * (size_t)(POOLED_H * POOLED_W);
    }
}

extern "C" void kernel_launch(void* const* d_in, const int* in_sizes, int n_in,
                              void* d_out, int out_size, void* d_ws, size_t ws_size,
                              hipStream_t stream) {
    const float* feat = (const float*)d_in[0];
    const float* rois = (const float*)d_in[1];
    float* out = (float*)d_out;
    const int n_rois = in_sizes[1] / 6;
    dim3 grid(n_rois * POOLED_H);
    dim3 block(256);
    rroi_align_kernel<<<grid, block, 0, stream>>>(feat, rois, out);
}